// MVFFeatureNet_19318762898073
// MI455X (gfx1250) — compile-verified
//
#include <hip/hip_runtime.h>
#include <math.h>

typedef __attribute__((ext_vector_type(2))) float v2f;
typedef __attribute__((ext_vector_type(8))) float v8f;
typedef __attribute__((ext_vector_type(4))) unsigned int u32x4;
typedef __attribute__((ext_vector_type(4))) int i32x4;
typedef __attribute__((ext_vector_type(8))) int i32x8;

#ifndef __has_builtin
#define __has_builtin(x) 0
#endif
#define HAVE_WMMA_F32 __has_builtin(__builtin_amdgcn_wmma_f32_16x16x4_f32)
#define HAVE_TDM      __has_builtin(__builtin_amdgcn_tensor_load_to_lds)

// ---- problem constants (from reference) ----
#define NB        4
#define PGX       752          // pillar grid x
#define PGY       752          // pillar grid y
#define CGX       1000         // cylinder grid phi
#define CGY       30           // cylinder grid z
#define NSEG_P    (NB*PGX*PGY) // 2,262,016
#define NSEG_C    (NB*CGX*CGY) // 120,000
#define NCH       128
#define KDIM      20
#define BN_EPS_F  0.001f

// ---- workspace layout (float offsets) ----
#define WS_PACC   0                       // 4*NSEG_P: {sx,sy,sz,cnt} per pillar seg
#define WS_CACC   (WS_PACC + 4*NSEG_P)    // 4*NSEG_C: {sphi,sz,srho,cnt} per cyl seg
#define WS_BNSUM  (WS_CACC + 4*NSEG_C)    // 128
#define WS_BNSQ   (WS_BNSUM + NCH)        // 128
#define WS_APARM  (WS_BNSQ + NCH)         // 128 (a = gamma*rsqrt(var+eps))
#define WS_BPARM  (WS_APARM + NCH)        // 128 (b = beta - mu*a)
#define WS_ZERO_N (WS_APARM)              // zero accumulators + BN stats

// -------------------- per-point geometry --------------------
struct PointGeom {
  float w;
  int   segP, segC;
  float x, y, z, e1, e2, phi, rho;
  float icPx, icPy, icC0, icC1;
};

__device__ __forceinline__ PointGeom computeGeom(const float* p6) {
  PointGeom g;
  float b = p6[0];
  g.x = p6[1]; g.y = p6[2]; g.z = p6[3]; g.e1 = p6[4]; g.e2 = p6[5];
  bool in = (g.x >= -75.2f) & (g.x < 75.2f) &
            (g.y >= -75.2f) & (g.y < 75.2f) &
            (g.z >= -2.0f)  & (g.z < 4.0f);
  g.w = in ? 1.0f : 0.0f;
  float cx = fminf(fmaxf((g.x + 75.2f) / 0.2f, 0.0f), 751.0f);
  float cy = fminf(fmaxf((g.y + 75.2f) / 0.2f, 0.0f), 751.0f);
  g.icPx = floorf(cx); g.icPy = floorf(cy);
  int bi = (int)b;
  g.segP = (bi * PGX + (int)g.icPx) * PGY + (int)g.icPy;
  g.phi = atan2f(g.y, g.x) * (180.0f / 3.14159265358979323846f);
  g.rho = sqrtf(g.x * g.x + g.y * g.y);
  float c0 = fminf(fmaxf((g.phi + 180.0f) / 0.36f, 0.0f), 999.0f);
  float c1 = fminf(fmaxf((g.z + 2.0f) / 0.2f, 0.0f), 29.0f);
  g.icC0 = floorf(c0); g.icC1 = floorf(c1);
  g.segC = (bi * CGX + (int)g.icC0) * CGY + (int)g.icC1;
  return g;
}

// -------------------- 20-feature builder (segment-mean gather) --------------------
__device__ __forceinline__ void computeFeats(const float* p6,
                                             const float* __restrict__ ws,
                                             float* __restrict__ f) {
  PointGeom g = computeGeom(p6);
  float4 pv = *(const float4*)(ws + WS_PACC + (size_t)g.segP * 4);  // 16B aligned
  float ip = 1.0f / fmaxf(pv.w, 1.0f);
  float mx = pv.x * ip, my = pv.y * ip, mz = pv.z * ip;
  float4 cv = *(const float4*)(ws + WS_CACC + (size_t)g.segC * 4);
  float ic = 1.0f / fmaxf(cv.w, 1.0f);
  float mphi = cv.x * ic, mzz = cv.y * ic, mrho = cv.z * ic;
  float w = g.w;
  f[0]  = g.x * w;  f[1] = g.y * w;  f[2] = g.z * w;  f[3] = g.e1 * w;  f[4] = g.e2 * w;
  f[5]  = (g.x - mx) * w;  f[6] = (g.y - my) * w;  f[7] = (g.z - mz) * w;
  f[8]  = (g.x - (g.icPx * 0.2f + 0.1f - 75.2f)) * w;
  f[9]  = (g.y - (g.icPy * 0.2f + 0.1f - 75.2f)) * w;
  f[10] = g.phi * w; f[11] = g.z * w; f[12] = g.rho * w; f[13] = g.e1 * w; f[14] = g.e2 * w;
  f[15] = (g.phi - mphi) * w; f[16] = (g.z - mzz) * w; f[17] = (g.rho - mrho) * w;
  f[18] = (g.phi - (g.icC0 * 0.36f + 0.18f - 180.0f)) * w;
  f[19] = (g.z   - (g.icC1 * 0.2f  + 0.1f  - 2.0f))   * w;
}

// -------------------- TDM helpers (gfx1250 Tensor Data Mover) --------------------
#if HAVE_TDM
__device__ __forceinline__ unsigned ldsOffsetOf(const void* p) {
  // flat LDS aperture: low 32 bits of the generic address are the LDS byte addr
  return (unsigned)(unsigned long long)(uintptr_t)p;
}

// 1-D tile: 96 contiguous f32 (one 16-point tile of the [N,6] array) -> LDS
__device__ __forceinline__ void tdm_load_tile(const float* gsrc, unsigned ldsAddr) {
  unsigned long long ga = (unsigned long long)(uintptr_t)gsrc;
  u32x4 g0;
  g0.x = 1u;                                               // count=1 user D#
  g0.y = ldsAddr;                                          // lds_addr (bytes)
  g0.z = (unsigned)(ga & 0xFFFFFFFFu);                     // global_addr[31:0]
  g0.w = (unsigned)((ga >> 32) & 0x1FFFFFFu) | 0x80000000u; // addr[56:32] | type=2
  i32x8 g1 = { 0x00020000,     // data_size = 4B, no multicast / iterate / pad
               (96 << 16),     // tensor_dim0[15:0] = 96
               0,              // tensor_dim0 hi / tensor_dim1 = 0
               (96 << 16),     // tile_dim0 = 96
               0,              // tile_dim1 = tile_dim2 = 0 (1-D)
               96,             // tensor_dim0_stride
               0, 0 };
  i32x4 zz4 = {0, 0, 0, 0};
  i32x8 zz8 = {0, 0, 0, 0, 0, 0, 0, 0};
  // clang-23 / therock signature: (g0, g1, g2, g3, g4, cpol) -- 6 args
  __builtin_amdgcn_tensor_load_to_lds(g0, g1, zz4, zz4, zz8, 0);
}

__device__ __forceinline__ void tdm_wait0() {
#if __has_builtin(__builtin_amdgcn_s_wait_tensorcnt)
  __builtin_amdgcn_s_wait_tensorcnt(0);
#else
  asm volatile("s_wait_tensorcnt 0x0" ::: "memory");
#endif
}
#endif

// -------------------- pass 1: zero workspace --------------------
__global__ __launch_bounds__(256) void zero_ws(float4* __restrict__ ws4, int n4) {
  int stride = gridDim.x * blockDim.x;
  for (int i = blockIdx.x * blockDim.x + threadIdx.x; i < n4; i += stride)
    ws4[i] = make_float4(0.f, 0.f, 0.f, 0.f);
}

// -------------------- pass 2: segment scatter-add --------------------
__global__ __launch_bounds__(256) void scatter(const float* __restrict__ pts,
                                               float* __restrict__ ws, int n) {
  int i = blockIdx.x * blockDim.x + threadIdx.x;
  if (i >= n) return;
  PointGeom g = computeGeom(pts + (size_t)i * 6);
  if (g.w != 0.0f) {   // w in {0,1}: skipping zero-weight adds is exact
    float* pa = ws + WS_PACC + (size_t)g.segP * 4;
    atomicAdd(pa + 0, g.x); atomicAdd(pa + 1, g.y);
    atomicAdd(pa + 2, g.z); atomicAdd(pa + 3, 1.0f);
    float* ca = ws + WS_CACC + (size_t)g.segC * 4;
    atomicAdd(ca + 0, g.phi); atomicAdd(ca + 1, g.z);
    atomicAdd(ca + 2, g.rho); atomicAdd(ca + 3, 1.0f);
  }
}

// -------------------- WMMA tile pipeline (shared by stats & apply passes) ----
// 256 threads = 8 wave32s; block = 16 points x 128 channels per tile.
// APPLY=false: accumulate per-channel sum/sumsq only (no h store).
// APPLY=true : recompute h, fuse BN+ReLU, single store of the final output.
template <bool APPLY>
__device__ __forceinline__ void tile_pipeline(const float* __restrict__ pts,
                                              const float* __restrict__ W1,
                                              float* __restrict__ out,
                                              float* __restrict__ ws, int nTiles) {
  __shared__ float Wlds[NCH * KDIM];    // 10 KB
  __shared__ float featLds[16 * KDIM];  // 1.25 KB
  __shared__ float sSum[NCH], sSq[NCH];
#if HAVE_TDM
  __shared__ float ptsLds[2][96];       // double-buffered TDM landing zone
#endif

  const int tid = threadIdx.x;
  for (int i = tid; i < NCH * KDIM; i += 256) Wlds[i] = W1[i];
  if (!APPLY && tid < NCH) { sSum[tid] = 0.0f; sSq[tid] = 0.0f; }

  const int wave = tid >> 5, lane = tid & 31;
  const int row = lane & 15, hi = lane >> 4;
  const int c0 = wave * 16, ch = c0 + row;

  float aCh = 0.0f, bCh = 0.0f;
  if (APPLY) { aCh = ws[WS_APARM + ch]; bCh = ws[WS_BPARM + ch]; }
  float accS = 0.0f, accQ = 0.0f;

  int t = blockIdx.x;
#if HAVE_TDM
  int buf = 0;
  if (t < nTiles && wave == 0)
    tdm_load_tile(pts + (size_t)t * 96, ldsOffsetOf(&ptsLds[0][0]));
#endif

  for (; t < nTiles; t += gridDim.x) {
    __syncthreads();  // featLds consumers from previous iteration are done
#if HAVE_TDM
    if (wave == 0) {
      tdm_wait0();                               // tile t landed in ptsLds[buf]
      int tn = t + gridDim.x;
      if (tn < nTiles)                           // prefetch next tile via TDM
        tdm_load_tile(pts + (size_t)tn * 96, ldsOffsetOf(&ptsLds[buf ^ 1][0]));
      if (lane < 16)
        computeFeats(&ptsLds[buf][lane * 6], ws, &featLds[lane * KDIM]);
    }
    buf ^= 1;
#else
    if (tid < 16)
      computeFeats(pts + (size_t)(t * 16 + tid) * 6, ws, &featLds[tid * KDIM]);
#endif
    __syncthreads();

    v8f c = {0.f, 0.f, 0.f, 0.f, 0.f, 0.f, 0.f, 0.f};
#if HAVE_WMMA_F32
    // A (16x4 f32): lanes 0-15 -> M=row, K={kb,kb+1}; lanes 16-31 -> K={kb+2,kb+3}
    // B (4x16 f32): same lane pattern, N=row, sourced from W1[c0+N][k]
    #pragma unroll
    for (int kk = 0; kk < 5; kk++) {
      const int kb = kk * 4 + 2 * hi;
      v2f a; a.x = featLds[row * KDIM + kb]; a.y = featLds[row * KDIM + kb + 1];
      v2f b; b.x = Wlds[ch * KDIM + kb];     b.y = Wlds[ch * KDIM + kb + 1];
      c = __builtin_amdgcn_wmma_f32_16x16x4_f32(false, a, false, b, (short)0, c,
                                                false, false);
    }
#else
    #pragma unroll
    for (int r = 0; r < 8; r++) {
      float s = 0.0f; const int m = r + 8 * hi;
      for (int k = 0; k < KDIM; k++) s += featLds[m * KDIM + k] * Wlds[ch * KDIM + k];
      c[r] = s;
    }
#endif
    // C/D layout: VGPR r -> M = r + 8*hi, N = ch
    if (APPLY) {
      // one 64-bit base per tile; stores use constant 24-bit immediate offsets
      float* op = out + (size_t)t * (16 * NCH) + (8 * hi) * NCH + ch;
      #pragma unroll
      for (int r = 0; r < 8; r++)
        op[r * NCH] = fmaxf(fmaf(c[r], aCh, bCh), 0.0f);
    } else {
      #pragma unroll
      for (int r = 0; r < 8; r++) { float v = c[r]; accS += v; accQ += v * v; }
    }
  }

  if (!APPLY) {
    atomicAdd(&sSum[ch], accS);   // ds_add_f32; hi=0/1 lanes share a channel
    atomicAdd(&sSq[ch], accQ);
    __syncthreads();
    if (tid < NCH) {
      atomicAdd(ws + WS_BNSUM + tid, sSum[tid]);
      atomicAdd(ws + WS_BNSQ  + tid, sSq[tid]);
    }
  }
}

__global__ __launch_bounds__(256) void gemm_stats(const float* __restrict__ pts,
                                                  const float* __restrict__ W1,
                                                  float* __restrict__ ws, int nTiles) {
  tile_pipeline<false>(pts, W1, nullptr, ws, nTiles);
}

__global__ __launch_bounds__(256) void gemm_apply(const float* __restrict__ pts,
                                                  const float* __restrict__ W1,
                                                  float* __restrict__ out,
                                                  float* __restrict__ ws, int nTiles) {
  tile_pipeline<true>(pts, W1, out, ws, nTiles);
}

// -------------------- pass 4: BN parameter finalize --------------------
__global__ void finalize(const float* __restrict__ gamma, const float* __restrict__ beta,
                         float* __restrict__ ws, float invN) {
  int c = threadIdx.x;
  if (c < NCH) {
    float mu  = ws[WS_BNSUM + c] * invN;
    float var = fmaxf(ws[WS_BNSQ + c] * invN - mu * mu, 0.0f);  // population var
    float rs  = rsqrtf(var + BN_EPS_F);
    float a   = gamma[c] * rs;
    ws[WS_APARM + c] = a;
    ws[WS_BPARM + c] = beta[c] - mu * a;
  }
}

// -------------------- launcher --------------------
extern "C" void kernel_launch(void* const* d_in, const int* in_sizes, int n_in,
                              void* d_out, int out_size, void* d_ws, size_t ws_size,
                              hipStream_t stream) {
  const float* pts   = (const float*)d_in[0];  // [N,6]: bidx,x,y,z,e1,e2
  const float* W1    = (const float*)d_in[1];  // [128,20]
  const float* gamma = (const float*)d_in[2];  // [128]
  const float* beta  = (const float*)d_in[3];  // [128]
  float* out = (float*)d_out;                  // [N,128]
  float* ws  = (float*)d_ws;                   // ~38.1 MB used

  const int n      = in_sizes[0] / 6;          // 1,000,000
  const int nTiles = n / 16;                   // 62,500 (exact)

  zero_ws<<<2048, 256, 0, stream>>>((float4*)ws, WS_ZERO_N / 4);
  scatter<<<(n + 255) / 256, 256, 0, stream>>>(pts, ws, n);
  gemm_stats<<<2048, 256, 0, stream>>>(pts, W1, ws, nTiles);
  finalize<<<1, 128, 0, stream>>>(gamma, beta, ws, 1.0f / (float)n);
  gemm_apply<<<2048, 256, 0, stream>>>(pts, W1, out, ws, nTiles);
}